// FlashExpressionAttentionLayer_34995393528237
// MI455X (gfx1250) — compile-verified
//
#include <hip/hip_runtime.h>
#include <hip/hip_bf16.h>

// ---------------------------------------------------------------------------
// FlashExpressionAttentionLayer for MI455X (gfx1250, wave32, WMMA f32 16x16x4)
//
// fused = gene@Wf_top + expr@Wf_bot + bf          (two accumulating GEMMs)
// Q = fused@Wq+bq ; K = fused@Wk+bk ; V = expr@Wv+bv
// attn: flash-style per (b,h,64-q-tile): S=QK^T*scale, p=exp(s-m)*mask,
//       l=sum p, O=sum p*V, out=O/(l*(1+eps))   (masked logits -> p == 0.0f)
// out = attnOut@Wo + bo
//
// Workspace (fp32): fused | Q | K | V | attnOut  = 5 * B*S*D = 40 MB.
//
// CDNA5 specifics used:
//  - v_wmma_f32_16x16x4_f32 (exact fp32 tensor path)
//  - global_load_async_to_lds_b128 + s_wait_asynccnt (ASYNCcnt DMA staging)
//  - pair-interleaved LDS so every WMMA operand is one aligned ds_load_b64
//  - sched_group_barrier pipelining + relaxed occupancy so DS latency hides
//    behind WMMA groups (one s_wait per WMMA group, not per WMMA pair)
//  - global_prefetch_b8 for the mask stream
// ---------------------------------------------------------------------------

typedef float v2f __attribute__((ext_vector_type(2)));
typedef float v8f __attribute__((ext_vector_type(8)));

#define B_     2
#define S_     2048
#define D_     512
#define H_     8
#define HD_    64
#define SCALE_ 0.125f     // HD^-0.5
#define EPS_   1e-8f

// sched_group_barrier masks
#define SG_WMMA   0x008
#define SG_DSREAD 0x100

__device__ __forceinline__ v8f vzero8() {
  v8f v;
#pragma unroll
  for (int i = 0; i < 8; ++i) v[i] = 0.0f;
  return v;
}

// D = A(16x4) * B(4x16) + C, fp32 exact.
__device__ __forceinline__ v8f wmma4(v2f a, v2f b, v8f c) {
  return __builtin_amdgcn_wmma_f32_16x16x4_f32(
      /*neg_a=*/false, a, /*neg_b=*/false, b,
      /*c_mod=*/(short)0, c, /*reuse_a=*/false, /*reuse_b=*/false);
}

// Memory -> LDS async DMA (16B per lane), tracked by ASYNCcnt.
__device__ __forceinline__ void async_copy16(void* lds_ptr, const void* gptr) {
  unsigned lds = (unsigned)(uintptr_t)lds_ptr;
  unsigned long long ga = (unsigned long long)(uintptr_t)gptr;
  asm volatile("global_load_async_to_lds_b128 %0, %1, off"
               :: "v"(lds), "v"(ga)
               : "memory");
}

__device__ __forceinline__ void wait_async_all() {
  asm volatile("s_wait_asynccnt 0x0" ::: "memory");
}

// ---------------------------------------------------------------------------
// Y[M x N] = X[M x K] @ W[K x N] (+bias) (+=Y if acc)
// grid = (M/128, N/128), block = 256 threads = 8 waves.
// Wave w: 16-row strip x full 128 cols -> 8 WMMA C tiles (64 accum VGPRs).
// ---------------------------------------------------------------------------
__global__ __launch_bounds__(256, 1) void gemm_bias_kernel(
    const float* __restrict__ X, const float* __restrict__ W,
    const float* __restrict__ bias, float* __restrict__ Y,
    int K, int N, int acc) {
  __shared__ float As[128][36];   // rows on distinct mult-of-4 banks
  __shared__ float Bs2[16][288];  // pair-interleaved; 288%64==32 -> halves split banks

  const int tid  = threadIdx.x;
  const int lane = tid & 31;
  const int wave = tid >> 5;
  const int m    = lane & 15;
  const int kh   = lane >> 4;
  const int row0 = blockIdx.x * 128;
  const int col0 = blockIdx.y * 128;

  v8f accv[8];
#pragma unroll
  for (int t = 0; t < 8; ++t) accv[t] = vzero8();

  for (int k0 = 0; k0 < K; k0 += 32) {
    // --- stage A via async DMA: 128x32 = 1024 float4, 4 per thread ---
#pragma unroll
    for (int it = 0; it < 4; ++it) {
      int idx = tid + it * 256;
      int r = idx >> 3, c4 = idx & 7;
      async_copy16(&As[r][c4 * 4],
                   &X[(size_t)(row0 + r) * (size_t)K + (size_t)(k0 + c4 * 4)]);
    }
    // --- stage B pair-interleaved: 16 k-pairs x 64 col-pairs, 4 per thread ---
#pragma unroll
    for (int it = 0; it < 4; ++it) {
      int idx = tid + it * 256;
      int r2 = idx >> 6, g = idx & 63;
      int n0 = g * 2;
      const float2 w0 = *reinterpret_cast<const float2*>(
          &W[(size_t)(k0 + 2 * r2) * (size_t)N + (size_t)(col0 + n0)]);
      const float2 w1 = *reinterpret_cast<const float2*>(
          &W[(size_t)(k0 + 2 * r2 + 1) * (size_t)N + (size_t)(col0 + n0)]);
      float4 v;
      v.x = w0.x; v.y = w1.x; v.z = w0.y; v.w = w1.y;
      *reinterpret_cast<float4*>(&Bs2[r2][n0 * 2]) = v;
    }
    wait_async_all();
    __syncthreads();

    // --- preload all 8 A fragments (merge to 4 ds_load_2addr_b64) ---
    v2f afr[8];
#pragma unroll
    for (int kk = 0; kk < 8; ++kk)
      afr[kk] = *reinterpret_cast<const v2f*>(&As[wave * 16 + m][kk * 4 + kh * 2]);

    // --- double-buffered B fragments ---
    v2f bcur[8], bnxt[8];
    {
      const float* brow = &Bs2[kh][m * 2];
#pragma unroll
      for (int t = 0; t < 8; ++t)
        bcur[t] = *reinterpret_cast<const v2f*>(brow + t * 32);
    }
#pragma unroll
    for (int kk = 0; kk < 8; ++kk) {
      if (kk < 7) {
        const float* brow = &Bs2[(kk + 1) * 2 + kh][m * 2];
#pragma unroll
        for (int t = 0; t < 8; ++t)
          bnxt[t] = *reinterpret_cast<const v2f*>(brow + t * 32);
      }
#pragma unroll
      for (int t = 0; t < 8; ++t)
        accv[t] = wmma4(afr[kk], bcur[t], accv[t]);
      if (kk < 7) {
#pragma unroll
        for (int t = 0; t < 8; ++t) bcur[t] = bnxt[t];
      }
    }

    // pin the pipeline: [8 DS][ (8 WMMA, 4 DS) x7 ][8 WMMA]
    __builtin_amdgcn_sched_group_barrier(SG_DSREAD, 8, 0);
#pragma unroll
    for (int kk = 0; kk < 8; ++kk) {
      __builtin_amdgcn_sched_group_barrier(SG_WMMA, 8, 0);
      if (kk < 7) __builtin_amdgcn_sched_group_barrier(SG_DSREAD, 4, 0);
    }
    __syncthreads();
  }

  // store: C row = rbase + i, col = col0 + t*16 + m
  const int rbase = row0 + wave * 16 + kh * 8;
#pragma unroll
  for (int t = 0; t < 8; ++t) {
    const int col = col0 + t * 16 + m;
    const float bv = bias ? bias[col] : 0.0f;
#pragma unroll
    for (int i = 0; i < 8; ++i) {
      const size_t off = (size_t)(rbase + i) * (size_t)N + (size_t)col;
      float v = accv[t][i] + bv;
      if (acc) v += Y[off];
      Y[off] = v;
    }
  }
}

// ---------------------------------------------------------------------------
// Flash attention with mask-folded renorm.
// grid = (S/64, H, B), block = 128 threads = 4 waves; wave owns 16 q-rows.
// ---------------------------------------------------------------------------
__global__ __launch_bounds__(128, 1) void flash_attn_kernel(
    const float* __restrict__ Qg, const float* __restrict__ Kg,
    const float* __restrict__ Vg, const float* __restrict__ Mg,
    float* __restrict__ Og) {
  __shared__ float Qs[64][68];     // q-tile (staged once, hoisted to regs)
  __shared__ float Ks[16][68];     // k-tile, row-major
  __shared__ float Vs2[8][160];    // v-tile pair-interleaved; 160%64==32
  __shared__ float Ps[4][16][20];  // per-wave P relayout

  const int tid  = threadIdx.x;
  const int lane = tid & 31;
  const int wave = tid >> 5;
  const int m    = lane & 15;
  const int kh   = lane >> 4;
  const int q0   = blockIdx.x * 64;
  const int h    = blockIdx.y;
  const int b    = blockIdx.z;

  // stage Q tile via async DMA: 64x64 = 1024 float4, 8 per thread
#pragma unroll
  for (int it = 0; it < 8; ++it) {
    int idx = tid + it * 128;
    int r = idx >> 4, c4 = idx & 15;
    async_copy16(&Qs[r][c4 * 4],
                 &Qg[(size_t)(b * S_ + q0 + r) * (size_t)D_ + (size_t)(h * HD_ + c4 * 4)]);
  }
  wait_async_all();
  __syncthreads();

  // hoist all 16 Q fragments out of the k-loop (invariant): 32 VGPRs
  v2f qf[16];
#pragma unroll
  for (int kk = 0; kk < 16; ++kk)
    qf[kk] = *reinterpret_cast<const v2f*>(&Qs[wave * 16 + m][kk * 4 + kh * 2]);

  float mrow[8], lrow[8];
#pragma unroll
  for (int i = 0; i < 8; ++i) { mrow[i] = -3.0e38f; lrow[i] = 0.0f; }
  v8f o[4];
#pragma unroll
  for (int t = 0; t < 4; ++t) o[t] = vzero8();

  const int qrow = q0 + wave * 16 + kh * 8;  // this lane's C-row base

  for (int k0 = 0; k0 < S_; k0 += 16) {
    __syncthreads();  // prior readers of Ks/Vs2 done
    // stage K via async DMA: 16x64 = 256 float4, 2 per thread
#pragma unroll
    for (int it = 0; it < 2; ++it) {
      int idx = tid + it * 128;
      int r = idx >> 4, c4 = idx & 15;
      async_copy16(&Ks[r][c4 * 4],
                   &Kg[(size_t)(b * S_ + k0 + r) * (size_t)D_ + (size_t)(h * HD_ + c4 * 4)]);
    }
    // stage V pair-interleaved: 8 k-pairs x 32 d-pairs = 256 tasks, 2/thread
#pragma unroll
    for (int it = 0; it < 2; ++it) {
      int idx = tid + it * 128;
      int r2 = idx >> 5, g = idx & 31;
      int d0 = g * 2;
      const float2 a0 = *reinterpret_cast<const float2*>(
          &Vg[(size_t)(b * S_ + k0 + 2 * r2) * (size_t)D_ + (size_t)(h * HD_ + d0)]);
      const float2 a1 = *reinterpret_cast<const float2*>(
          &Vg[(size_t)(b * S_ + k0 + 2 * r2 + 1) * (size_t)D_ + (size_t)(h * HD_ + d0)]);
      float4 v;
      v.x = a0.x; v.y = a1.x; v.z = a0.y; v.w = a1.y;
      *reinterpret_cast<float4*>(&Vs2[r2][d0 * 2]) = v;
    }
    wait_async_all();
    __syncthreads();

    // prefetch next mask tile (global_prefetch_b8)
    if (k0 + 16 < S_) {
      __builtin_prefetch(&Mg[((size_t)b * S_ + (size_t)qrow) * (size_t)S_ +
                             (size_t)(k0 + 16 + m)], 0, 3);
    }

    // load ALL 16 K fragments, then 16 straight WMMAs (one wait per group);
    // two accumulator chains so dependent XDL latency doesn't serialize.
    v2f kf[16];
#pragma unroll
    for (int kk = 0; kk < 16; ++kk)
      kf[kk] = *reinterpret_cast<const v2f*>(&Ks[m][kk * 4 + kh * 2]);
    v8f s0 = vzero8(), s1 = vzero8();
#pragma unroll
    for (int kk = 0; kk < 8; ++kk) {
      s0 = wmma4(qf[2 * kk + 0], kf[2 * kk + 0], s0);
      s1 = wmma4(qf[2 * kk + 1], kf[2 * kk + 1], s1);
    }
    v8f s;
#pragma unroll
    for (int i = 0; i < 8; ++i) s[i] = s0[i] + s1[i];

    // mask + online softmax; lane holds rows (qrow+i), i=0..7, col = m
#pragma unroll
    for (int i = 0; i < 8; ++i) {
      const float mk =
          Mg[((size_t)b * S_ + (size_t)(qrow + i)) * (size_t)S_ + (size_t)(k0 + m)];
      const float x = (mk != 0.0f) ? s[i] * SCALE_ : -3.0e38f;
      float rmax = x;
      rmax = fmaxf(rmax, __shfl_xor(rmax, 1, 32));
      rmax = fmaxf(rmax, __shfl_xor(rmax, 2, 32));
      rmax = fmaxf(rmax, __shfl_xor(rmax, 4, 32));
      rmax = fmaxf(rmax, __shfl_xor(rmax, 8, 32));
      const float mnew = fmaxf(mrow[i], rmax);
      const float p = (x > -1.0e38f) ? __expf(x - mnew) : 0.0f;
      float psum = p;
      psum += __shfl_xor(psum, 1, 32);
      psum += __shfl_xor(psum, 2, 32);
      psum += __shfl_xor(psum, 4, 32);
      psum += __shfl_xor(psum, 8, 32);
      const float alpha = __expf(mrow[i] - mnew);
      lrow[i] = lrow[i] * alpha + psum;
      mrow[i] = mnew;
#pragma unroll
      for (int t = 0; t < 4; ++t) o[t][i] *= alpha;
      Ps[wave][kh * 8 + i][m] = p;  // C-layout -> LDS for A-layout reread
    }
    __syncthreads();  // P (cross-lane) visibility

    // load ALL P + V fragments, then 16 straight WMMAs (4 independent chains)
    v2f pf[4], vf[16];
#pragma unroll
    for (int kk = 0; kk < 4; ++kk)
      pf[kk] = *reinterpret_cast<const v2f*>(&Ps[wave][m][kk * 4 + kh * 2]);
#pragma unroll
    for (int kk = 0; kk < 4; ++kk) {
      const float* vrow = &Vs2[kk * 2 + kh][m * 2];
#pragma unroll
      for (int t = 0; t < 4; ++t)
        vf[kk * 4 + t] = *reinterpret_cast<const v2f*>(vrow + t * 32);
    }
#pragma unroll
    for (int kk = 0; kk < 4; ++kk) {
#pragma unroll
      for (int t = 0; t < 4; ++t)
        o[t] = wmma4(pf[kk], vf[kk * 4 + t], o[t]);
    }
  }

  // out = O / (l*(1+eps)); masked exp's are exact 0, so l == softmax mass * M
#pragma unroll
  for (int i = 0; i < 8; ++i) {
    const float l = lrow[i];
    const float r = (l > 0.0f) ? 1.0f / (l * (1.0f + EPS_)) : 0.0f;
    const size_t base = (size_t)(b * S_ + qrow + i) * (size_t)D_ + (size_t)(h * HD_);
#pragma unroll
    for (int t = 0; t < 4; ++t) {
      Og[base + t * 16 + m] = o[t][i] * r;
    }
  }
}

// ---------------------------------------------------------------------------
extern "C" void kernel_launch(void* const* d_in, const int* in_sizes, int n_in,
                              void* d_out, int out_size, void* d_ws, size_t ws_size,
                              hipStream_t stream) {
  const float* gene = (const float*)d_in[0];
  const float* expr = (const float*)d_in[1];
  const float* Mm   = (const float*)d_in[2];
  const float* Wf   = (const float*)d_in[3];
  const float* bf   = (const float*)d_in[4];
  const float* Wq   = (const float*)d_in[5];
  const float* bq   = (const float*)d_in[6];
  const float* Wk   = (const float*)d_in[7];
  const float* bk   = (const float*)d_in[8];
  const float* Wv   = (const float*)d_in[9];
  const float* bv   = (const float*)d_in[10];
  const float* Wo   = (const float*)d_in[11];
  const float* bo   = (const float*)d_in[12];
  float* out = (float*)d_out;

  float* ws = (float*)d_ws;
  const size_t RD = (size_t)B_ * S_ * D_;  // 2,097,152 floats
  float* fused = ws;
  float* Qw    = ws + RD;
  float* Kw    = ws + 2 * RD;
  float* Vw    = ws + 3 * RD;
  float* AO    = ws + 4 * RD;

  const dim3 gGrid((B_ * S_) / 128, D_ / 128);  // 32 x 4
  // fused = gene@Wf[0:512] + bf ; fused += expr@Wf[512:1024]
  gemm_bias_kernel<<<gGrid, 256, 0, stream>>>(gene, Wf,             bf,      fused, D_, D_, 0);
  gemm_bias_kernel<<<gGrid, 256, 0, stream>>>(expr, Wf + 512 * 512, nullptr, fused, D_, D_, 1);
  // Q, K, V projections
  gemm_bias_kernel<<<gGrid, 256, 0, stream>>>(fused, Wq, bq, Qw, D_, D_, 0);
  gemm_bias_kernel<<<gGrid, 256, 0, stream>>>(fused, Wk, bk, Kw, D_, D_, 0);
  gemm_bias_kernel<<<gGrid, 256, 0, stream>>>(expr,  Wv, bv, Vw, D_, D_, 0);
  // fused masked attention
  const dim3 aGrid(S_ / 64, H_, B_);
  flash_attn_kernel<<<aGrid, 128, 0, stream>>>(Qw, Kw, Vw, Mm, AO);
  // output projection
  gemm_bias_kernel<<<gGrid, 256, 0, stream>>>(AO, Wo, bo, out, D_, D_, 0);
}